// ConcatenateSumOperation1_48773648613703
// MI455X (gfx1250) — compile-verified
//
#include <hip/hip_runtime.h>

// 2-VGPR f32 operand (16x4 A / 4x16 B) and 8-VGPR f32 accumulator (16x16 C/D)
typedef __attribute__((ext_vector_type(2))) float v2f;
typedef __attribute__((ext_vector_type(8))) float v8f;

// Explicit global-address-space pointers so the backend emits global_load_b32
// (LOADcnt only) instead of flat_load_b32 (LOADcnt + DScnt + aperture check).
typedef const float __attribute__((address_space(1)))* gcfloat_p;
typedef float __attribute__((address_space(1)))* gfloat_p;

// out[b, i, d] = sum_l x_i[b, l, d]
// One block per (tensor i, batch b). 8 waves; wave w owns feature columns
// [16*w, 16*w+16). Row-sum done with V_WMMA_F32_16X16X4_F32 and an all-ones
// A matrix: each WMMA folds a 4(seq) x 16(feat) tile into the accumulator.
// With ones-A the result is invariant to the K-row <-> (VGPR, lane-half)
// assignment inside B; only the column->lane mapping (lane & 15) matters.
__global__ __launch_bounds__(256)
void sum_seq_cat_wmma_kernel(const float* __restrict__ x0, const float* __restrict__ x1,
                             const float* __restrict__ x2, const float* __restrict__ x3,
                             const float* __restrict__ x4, const float* __restrict__ x5,
                             const float* __restrict__ x6, const float* __restrict__ x7,
                             float* __restrict__ out)
{
    constexpr int D = 128;   // feature dim
    constexpr int B = 512;   // batch

    const int pair = blockIdx.x;       // 0..4095
    const int i    = pair >> 9;        // tensor index 0..7 (pair / 512)
    const int b    = pair & (B - 1);   // batch index
    const int L    = 64 * (i + 1);     // ragged seq length

    const float* xs[8] = {x0, x1, x2, x3, x4, x5, x6, x7};
    // generic -> global addrspacecast: all candidates are global kernel args
    gcfloat_p x = (gcfloat_p)xs[i];

    const int tid    = threadIdx.x;
    const int wave   = tid >> 5;          // 0..7 -> feature tile
    const int lane   = tid & 31;
    const int d0     = wave << 4;         // column tile base
    const int col    = lane & 15;         // B-matrix column for this lane
    const int rowoff = (lane >> 4) << 1;  // lanes 0-15 -> rows +0/+1, lanes 16-31 -> +2/+3

    // &x[b][rowoff][d0 + col]
    gcfloat_p base = x + (size_t)b * L * D + (size_t)rowoff * D + (d0 + col);

    const v2f ones = {1.0f, 1.0f};   // A = ones(16x4), layout-independent
    v8f c0 = {}; v8f c1 = {}; v8f c2 = {}; v8f c3 = {};

    // 16 sequence rows per iteration: 4 independent WMMA accumulation chains,
    // 8 outstanding b32 loads (1 KB fresh data per wave per iteration).
    for (int l = 0; l < L; l += 16) {
        gcfloat_p p = base + (size_t)l * D;
        v2f b0, b1, b2, b3;
        b0.x = p[0 * D];  b0.y = p[1 * D];    // rows l+rowoff(+1)
        b1.x = p[4 * D];  b1.y = p[5 * D];    // rows l+4+rowoff(+1)
        b2.x = p[8 * D];  b2.y = p[9 * D];    // rows l+8+rowoff(+1)
        b3.x = p[12 * D]; b3.y = p[13 * D];   // rows l+12+rowoff(+1)

        c0 = __builtin_amdgcn_wmma_f32_16x16x4_f32(false, ones, false, b0, (short)0, c0, false, false);
        c1 = __builtin_amdgcn_wmma_f32_16x16x4_f32(false, ones, false, b1, (short)0, c1, false, false);
        c2 = __builtin_amdgcn_wmma_f32_16x16x4_f32(false, ones, false, b2, (short)0, c2, false, false);
        c3 = __builtin_amdgcn_wmma_f32_16x16x4_f32(false, ones, false, b3, (short)0, c3, false, false);
    }

    // D row M=0 lives in accumulator VGPR0, lanes 0-15 (N = lane).
    if (lane < 16) {
        float s = (c0[0] + c1[0]) + (c2[0] + c3[0]);
        gfloat_p o = (gfloat_p)out;
        o[(size_t)(b * 8 + i) * D + (d0 + col)] = s;
    }
}

extern "C" void kernel_launch(void* const* d_in, const int* in_sizes, int n_in,
                              void* d_out, int out_size, void* d_ws, size_t ws_size,
                              hipStream_t stream) {
    (void)in_sizes; (void)n_in; (void)d_ws; (void)ws_size; (void)out_size;

    const float* x0 = (const float*)d_in[0];
    const float* x1 = (const float*)d_in[1];
    const float* x2 = (const float*)d_in[2];
    const float* x3 = (const float*)d_in[3];
    const float* x4 = (const float*)d_in[4];
    const float* x5 = (const float*)d_in[5];
    const float* x6 = (const float*)d_in[6];
    const float* x7 = (const float*)d_in[7];
    float* out = (float*)d_out;

    dim3 grid(8 * 512);   // one block per (tensor, batch)
    dim3 block(256);      // 8 waves (wave32)
    hipLaunchKernelGGL(sum_seq_cat_wmma_kernel, grid, block, 0, stream,
                       x0, x1, x2, x3, x4, x5, x6, x7, out);
}